// RGB_Net_22342419873903
// MI455X (gfx1250) — compile-verified
//
#include <hip/hip_runtime.h>

// ---------------------------------------------------------------------------
// RGB_Net scatter + bilinear-gather + sparse blend, MI455X (gfx1250, wave32)
//
// out[b,c,x,y] = 0 unless cell (x,y) was hit by a scattered point p:
//   out = d0*bilin(c3[b,c]) + d1*bilin(c4[b,c]) + d2*bilin(c5[b,c])
// with sample coords px*(W-1), py*(H-1)  (align_corners=True) and
// d_m = dynamic_img[b,m,x,y].
//
// Store-bandwidth bound (256 MB output, >92% zeros): use TH=NT b128 stores.
// No matmul structure exists -> WMMA intentionally not used.
// ---------------------------------------------------------------------------

typedef float f4 __attribute__((ext_vector_type(4)));
typedef int   i4 __attribute__((ext_vector_type(4)));

#define XSZ 512
#define YSZ 512
#define CCH 64

// gfx1250 async LDS->global store path (probe: builtin exists on this
// toolchain; signature wants v4i32 addrspace(1)* / addrspace(3)* pointers).
#if defined(__gfx1250__) && __has_builtin(__builtin_amdgcn_global_store_async_from_lds_b128)
#define USE_ASYNC_FILL 1
typedef __attribute__((address_space(1))) i4 gi4_t;   // global v4i32
typedef __attribute__((address_space(3))) i4 li4_t;   // LDS v4i32
#else
#define USE_ASYNC_FILL 0
#endif

// ---------------------------------------------------------------------------
// K1: fill winner array with -1 (4 MB). Uses the CDNA5 async
// store-from-LDS path: -1'ed LDS block is streamed to memory by the async
// unit, tracked with ASYNCcnt.
// ---------------------------------------------------------------------------
__global__ __launch_bounds__(256) void k_init_win(int* __restrict__ win, int n /* ints, mult of 4 */) {
  const int stride = gridDim.x * blockDim.x;
  const int t = blockIdx.x * blockDim.x + threadIdx.x;
#if USE_ASYNC_FILL
  __shared__ int lbuf[256 * 4];
  i4 m1 = { -1, -1, -1, -1 };
  ((i4*)lbuf)[threadIdx.x] = m1;
  __syncthreads();
  void* lsrc = (void*)&lbuf[threadIdx.x * 4];
  for (int i = t; i * 4 < n; i += stride) {
    __builtin_amdgcn_global_store_async_from_lds_b128(
        (gi4_t*)(win + (size_t)i * 4),
        (li4_t*)lsrc,
        /*offset=*/0, /*cpol=*/0);
  }
#if __has_builtin(__builtin_amdgcn_s_wait_asynccnt)
  __builtin_amdgcn_s_wait_asynccnt(0);
#else
  asm volatile("s_wait_asynccnt 0" ::: "memory");
#endif
#else
  i4 m1 = { -1, -1, -1, -1 };
  for (int i = t; i * 4 < n; i += stride) {
    __builtin_nontemporal_store(m1, (i4*)(win + (size_t)i * 4));
  }
#endif
}

// ---------------------------------------------------------------------------
// K2: scatter points. Deterministic last-write-wins via atomicMax on point
// index (matches sequential JAX scatter where the largest p wins).
// ---------------------------------------------------------------------------
__global__ __launch_bounds__(256) void k_scatter(const int* __restrict__ coors,
                                                 const int* __restrict__ contains,
                                                 int* __restrict__ win, int BP, int P) {
  int t = blockIdx.x * blockDim.x + threadIdx.x;
  if (t >= BP) return;
  if (contains[t] <= 0) return;
  int b = t / P;
  int p = t - b * P;
  int cx = coors[(size_t)t * 3 + 1];
  int cy = coors[(size_t)t * 3 + 2];
  if ((unsigned)cx >= (unsigned)XSZ || (unsigned)cy >= (unsigned)YSZ) return;  // mode='drop'
  atomicMax(&win[((size_t)b * XSZ + cx) * YSZ + cy], p);
}

// ---------------------------------------------------------------------------
// Bilinear sample, reference-exact (align_corners=True, zeros padding,
// per-tap validity, clamp-then-gather).
// ---------------------------------------------------------------------------
__device__ __forceinline__ float sample_one(const float* __restrict__ img, int H, int W,
                                            float gx, float gy) {
  float fx = (gx + 1.0f) * 0.5f * (float)(W - 1);
  float fy = (gy + 1.0f) * 0.5f * (float)(H - 1);
  float x0 = floorf(fx), y0 = floorf(fy);
  float wx1 = fx - x0, wy1 = fy - y0;
  float wx0 = 1.0f - wx1, wy0 = 1.0f - wy1;
  float acc = 0.0f;
#pragma unroll
  for (int dy = 0; dy < 2; ++dy) {
#pragma unroll
    for (int dx = 0; dx < 2; ++dx) {
      float xi = x0 + (float)dx;
      float yi = y0 + (float)dy;
      bool valid = (xi >= 0.0f) && (xi <= (float)(W - 1)) &&
                   (yi >= 0.0f) && (yi <= (float)(H - 1));
      int xc = (int)fminf(fmaxf(xi, 0.0f), (float)(W - 1));
      int yc = (int)fminf(fmaxf(yi, 0.0f), (float)(H - 1));
      float w = (dx ? wx1 : wx0) * (dy ? wy1 : wy0);
      acc += img[yc * W + xc] * (valid ? w : 0.0f);
    }
  }
  return acc;
}

// ---------------------------------------------------------------------------
// K3: produce output. Thread t handles 4 consecutive y of one (b,c,x) row:
// wave of 32 lanes writes 512 contiguous bytes (coalesced b128 NT stores).
// Fast path (all 4 cells empty, ~92% of cells): single NT zero store.
// ---------------------------------------------------------------------------
__global__ __launch_bounds__(256) void k_out(const float* __restrict__ c3,
                                             const float* __restrict__ c4,
                                             const float* __restrict__ c5,
                                             const float* __restrict__ dyn,
                                             const float* __restrict__ pnts,
                                             const int* __restrict__ win,
                                             float* __restrict__ out, int P) {
  int i = blockIdx.x * blockDim.x + threadIdx.x;  // over B*64*512*128
  int yg = i & 127;          // y-group (4 y's each)
  int x  = (i >> 7) & 511;
  int c  = (i >> 16) & 63;
  int b  = i >> 22;
  int y0 = yg * 4;

  i4 w4 = *(const i4*)&win[((size_t)b * XSZ + x) * YSZ + y0];
  f4 r = { 0.0f, 0.0f, 0.0f, 0.0f };

  int mx = max(max(w4.x, w4.y), max(w4.z, w4.w));
  if (mx >= 0) {
    const float* b3 = c3 + (size_t)(b * CCH + c) * (48 * 160);
    const float* b4 = c4 + (size_t)(b * CCH + c) * (24 * 80);
    const float* b5 = c5 + (size_t)(b * CCH + c) * (12 * 40);
    size_t dbase = (((size_t)b * 3) * XSZ + x) * YSZ + y0;
    const size_t dstride = (size_t)XSZ * YSZ;
#pragma unroll
    for (int j = 0; j < 4; ++j) {
      int p = w4[j];
      if (p >= 0) {
        float px = pnts[((size_t)b * P + p) * 2 + 0];
        float py = pnts[((size_t)b * P + p) * 2 + 1];
        float gx = px * 2.0f - 1.0f;   // reference: pnts*2-1
        float gy = py * 2.0f - 1.0f;
        float v3 = sample_one(b3, 48, 160, gx, gy);
        float v4 = sample_one(b4, 24, 80, gx, gy);
        float v5 = sample_one(b5, 12, 40, gx, gy);
        float d0 = dyn[dbase + j];
        float d1 = dyn[dbase + j + dstride];
        float d2 = dyn[dbase + j + 2 * dstride];
        r[j] = v3 * d0 + v4 * d1 + v5 * d2;
      }
    }
  }
  // flat out index: ((b*64+c)*512+x)*512 + y0 == i*4 by construction
  __builtin_nontemporal_store(r, (f4*)(out + (size_t)i * 4));
}

// ---------------------------------------------------------------------------
// Launcher. Inputs (setup_inputs order):
//   0:c3 [B,64,48,160] f32   1:c4 [B,64,24,80]   2:c5 [B,64,12,40]
//   3:dynamic_img [B,3,512,512]  4:pillar_img_pts [B,P,2] f32
//   5:rgb_coors [B,P,3] i32      6:contains_rgb [B,P] i32
// Output: [B,64,512,512] f32. Workspace: B*512*512 ints (4 MB for B=4).
// ---------------------------------------------------------------------------
extern "C" void kernel_launch(void* const* d_in, const int* in_sizes, int n_in,
                              void* d_out, int out_size, void* d_ws, size_t ws_size,
                              hipStream_t stream) {
  const float* c3   = (const float*)d_in[0];
  const float* c4   = (const float*)d_in[1];
  const float* c5   = (const float*)d_in[2];
  const float* dyn  = (const float*)d_in[3];
  const float* pnts = (const float*)d_in[4];
  const int* coors    = (const int*)d_in[5];
  const int* contains = (const int*)d_in[6];

  const int B = in_sizes[0] / (CCH * 48 * 160);   // 4
  const int P = in_sizes[6] / B;                  // 20000

  int* win = (int*)d_ws;                          // B*512*512 ints
  const int nwin = B * XSZ * YSZ;

  k_init_win<<<512, 256, 0, stream>>>(win, nwin);

  const int BP = B * P;
  k_scatter<<<(BP + 255) / 256, 256, 0, stream>>>(coors, contains, win, BP, P);

  const long long total4 = (long long)B * CCH * XSZ * (YSZ / 4);
  k_out<<<(unsigned)(total4 / 256), 256, 0, stream>>>(c3, c4, c5, dyn, pnts, win,
                                                      (float*)d_out, P);
}